// AWingLoss_51591147159746
// MI455X (gfx1250) — compile-verified
//
#include <hip/hip_runtime.h>
#include <stdint.h>

// Problem constants (from reference): B=32, C=68, H=W=128
#define WIDTH      128
#define TILE_H     16
#define PLANES     (32 * 68)
#define SLABS      (WIDTH / TILE_H)          // 8
#define NBLOCKS    (PLANES * SLABS)          // 17408
#define NTOT       (32ll * 68 * 128 * 128)   // 35,651,584
#define LDS_STRIDE 132                       // 128 + pad, multiple of 4 floats (16B)

typedef __attribute__((ext_vector_type(2))) float v2f;
typedef __attribute__((ext_vector_type(4))) float v4f;
typedef __attribute__((ext_vector_type(8))) float v8f;

__device__ __forceinline__ float fmax3(float a, float b, float c) {
    return fmaxf(fmaxf(a, b), c);
}

// ---------------------------------------------------------------------------
// Kernel 1: per-block partial sums of masked AWing loss.
// Each block = one 16-row slab of one (b,c) 128x128 plane. Target rows
// [h0-1, h0+16] (edge-clamped) are staged into LDS with CDNA5 async copies;
// prediction streams through with non-temporal b128 loads issued BEFORE the
// async wait so HBM latency overlaps the LDS staging.
// ---------------------------------------------------------------------------
__global__ __launch_bounds__(256) void awing_partial_kernel(
    const float* __restrict__ pred,
    const float* __restrict__ tgt,
    float* __restrict__ partial)
{
    __shared__ float tile[(TILE_H + 2) * LDS_STRIDE];
    __shared__ float wsum[8];

    const int tid   = threadIdx.x;
    const int lane  = tid & 31;
    const int wave  = tid >> 5;
    const int plane = blockIdx.x >> 3;   // 0..2175
    const int slab  = blockIdx.x & 7;    // 0..7
    const int h0    = slab * TILE_H;

    const float* tp = tgt + (size_t)plane * (WIDTH * WIDTH);

    // ---- Stage target halo tile into LDS via async global->LDS DMA path ----
    // One wave-wide b128 per row: 32 lanes x 16B = 128 floats = one full row.
    for (int r = wave; r < TILE_H + 2; r += 8) {
        int gr = h0 - 1 + r;                       // symmetric pad == edge clamp for pad=1
        gr = gr < 0 ? 0 : (gr > WIDTH - 1 ? WIDTH - 1 : gr);
        const float* src = tp + (size_t)gr * WIDTH + lane * 4;
        uint32_t dst = (uint32_t)(uintptr_t)(&tile[r * LDS_STRIDE + lane * 4]);
        asm volatile("global_load_async_to_lds_b128 %0, %1, off"
                     :: "v"(dst), "v"(src)
                     : "memory");
    }

    // ---- Issue prediction loads NOW (overlap with async wait + barrier) ----
    const int lr = tid >> 4;             // local row 0..15
    const int c0 = (tid & 15) * 8;       // column base 0..120
    const float* pp = pred + (size_t)plane * (WIDTH * WIDTH)
                           + (size_t)(h0 + lr) * WIDTH + c0;
    v4f p0 = __builtin_nontemporal_load((const v4f*)(pp));       // zero reuse -> NT
    v4f p1 = __builtin_nontemporal_load((const v4f*)(pp + 4));

    asm volatile("s_wait_asynccnt 0x0" ::: "memory");
    __syncthreads();

    // Pull the 3x10 target neighborhood from LDS (tile row 0 == global h0-1).
    float v[3][10];
    #pragma unroll
    for (int rr = 0; rr < 3; ++rr) {
        const float* row = &tile[(lr + rr) * LDS_STRIDE];
        const int cl = (c0 == 0) ? 0 : c0 - 1;
        const int cr = (c0 == WIDTH - 8) ? WIDTH - 1 : c0 + 8;
        v[rr][0] = row[cl];
        #pragma unroll
        for (int j = 0; j < 8; ++j) v[rr][1 + j] = row[c0 + j];
        v[rr][9] = row[cr];
    }

    const float LN2 = 0.6931471805599453f;
    float sum = 0.0f;
    #pragma unroll
    for (int j = 0; j < 8; ++j) {
        const float t  = v[1][j + 1];
        const float pj = (j < 4) ? p0[j] : p1[j - 4];
        const float dil = fmax3(fmax3(v[0][j], v[0][j + 1], v[0][j + 2]),
                                fmax3(v[1][j], v[1][j + 1], v[1][j + 2]),
                                fmax3(v[2][j], v[2][j + 1], v[2][j + 2]));
        const float e  = 2.1f - t;                          // alpha - y  in [1.1, 2.1]
        const float pt = __builtin_amdgcn_exp2f(-e);        // (0.5)^(alpha-y) = 2^-e
        const float A  = 28.0f * e * pt * __builtin_amdgcn_rcpf(1.0f + pt); // 14*2pt*e/(1+pt)
        const float l1pt = LN2 * __builtin_amdgcn_logf(1.0f + pt);          // log1p(pt)
        const float Cc = 0.5f * A - 14.0f * l1pt;
        const float d  = fabsf(pj - t);
        // d^e = exp2(e*log2 d); d==0 -> log2=-inf -> exp2->0 -> log1p->0 (correct)
        const float dp = __builtin_amdgcn_exp2f(e * __builtin_amdgcn_logf(d));
        const float ls = 14.0f * LN2 * __builtin_amdgcn_logf(1.0f + dp);
        const float ll = A * d - Cc;
        float loss = (d < 0.5f) ? ls : ll;
        loss *= (dil > 0.2f) ? 11.0f : 1.0f;                // MASK_W + 1 or 1
        sum += loss;
    }

    // ---- Cross-lane fold using the matrix pipe (EXEC is all-ones here) ----
    // A = ones(16x4), B holds the 32 lane partials (other B slots zero):
    //   D[0][n] = column-sum(B);  sum_n D[0][n] = sum of ALL B slots = wave total.
    v2f a; a[0] = 1.0f; a[1] = 1.0f;
    v2f b; b[0] = sum;  b[1] = 0.0f;
    v8f c = {};
    c = __builtin_amdgcn_wmma_f32_16x16x4_f32(
            /*neg_a=*/false, a, /*neg_b=*/false, b,
            /*c_mod=*/(short)0, c, /*reuse_a=*/false, /*reuse_b=*/false);
    float s = c[0];                       // lane n: D[row][n], rows identical
    s += __shfl_xor(s, 8, 32);            // fold the 16 column sums
    s += __shfl_xor(s, 4, 32);
    s += __shfl_xor(s, 2, 32);
    s += __shfl_xor(s, 1, 32);

    if (lane == 0) wsum[wave] = s;
    __syncthreads();
    if (tid == 0) {
        float bs = 0.0f;
        #pragma unroll
        for (int i = 0; i < 8; ++i) bs += wsum[i];
        partial[blockIdx.x] = bs;
    }
}

// ---------------------------------------------------------------------------
// Kernel 2: deterministic final reduction (fixed order, f64 accumulation).
// NBLOCKS is a multiple of 1024, so the float4 strided loop is exact.
// ---------------------------------------------------------------------------
__global__ __launch_bounds__(256) void awing_final_kernel(
    const float* __restrict__ partial, float* __restrict__ out, int n4)
{
    __shared__ double wsum[8];
    const v4f* p4 = (const v4f*)partial;
    double acc = 0.0;
    for (int i = threadIdx.x; i < n4; i += 256) {
        v4f v = p4[i];
        acc += (double)v[0] + (double)v[1] + (double)v[2] + (double)v[3];
    }
    for (int off = 16; off; off >>= 1) acc += __shfl_xor(acc, off, 32);
    const int lane = threadIdx.x & 31;
    const int wave = threadIdx.x >> 5;
    if (lane == 0) wsum[wave] = acc;
    __syncthreads();
    if (threadIdx.x == 0) {
        double tot = 0.0;
        #pragma unroll
        for (int i = 0; i < 8; ++i) tot += wsum[i];
        out[0] = (float)(tot / (double)NTOT);
    }
}

extern "C" void kernel_launch(void* const* d_in, const int* in_sizes, int n_in,
                              void* d_out, int out_size, void* d_ws, size_t ws_size,
                              hipStream_t stream) {
    const float* pred = (const float*)d_in[0];
    const float* tgt  = (const float*)d_in[1];
    float* out        = (float*)d_out;
    float* partial    = (float*)d_ws;   // NBLOCKS floats = 68 KB scratch

    awing_partial_kernel<<<NBLOCKS, 256, 0, stream>>>(pred, tgt, partial);
    awing_final_kernel<<<1, 256, 0, stream>>>(partial, out, NBLOCKS / 4);
}